// ToRGB_17755394801862
// MI455X (gfx1250) — compile-verified
//
#include <hip/hip_runtime.h>
#include <hip/hip_bf16.h>
#include <math.h>

// -------- problem constants (from reference setup_inputs) --------
#define NB     8          // batch
#define CIN    512        // input channels
#define HOUT   256
#define WOUT   256
#define HW     (HOUT * WOUT)      // 65536 pixels per (b, channel) plane
#define HS     128                // skip spatial
#define LDSS   516                // padded LDS row stride (floats): even (b64-aligned) + bank-spread

typedef __attribute__((ext_vector_type(2))) float v2f;
typedef __attribute__((ext_vector_type(8))) float v8f;

// =================================================================
// Kernel 1: upfirdn2d(skip, k=outer([1,3,3,1])^2/64 * 4, up=2, pad=(2,2))
// One thread per output element; writes the upsampled value into d_out.
// =================================================================
__global__ __launch_bounds__(256) void upsample_blur_kernel(
    const float* __restrict__ skip, float* __restrict__ out) {
  int idx = blockIdx.x * 256 + threadIdx.x;        // < NB*3*HW
  int xo = idx & (WOUT - 1);
  int yo = (idx >> 8) & (HOUT - 1);
  int bo = idx >> 16;                              // b*3 + o
  const float k1f[4] = {1.f, 3.f, 3.f, 1.f};
  const float* sp = skip + (size_t)bo * (HS * HS);
  float acc = 0.f;
#pragma unroll
  for (int i = 0; i < 4; ++i) {
    int t = yo - 2 + i;
    if (t & 1) continue;                           // zero-inserted row
    int u = t >> 1;
    if ((unsigned)u >= (unsigned)HS) continue;     // zero padding
    float wy = k1f[3 - i];                         // flipped kernel
    const float* rowp = sp + u * HS;
#pragma unroll
    for (int j = 0; j < 4; ++j) {
      int s = xo - 2 + j;
      if (s & 1) continue;
      int vc = s >> 1;
      if ((unsigned)vc >= (unsigned)HS) continue;
      acc = fmaf(wy * k1f[3 - j], rowp[vc], acc);
    }
  }
  out[idx] = acc * (1.f / 16.f);                   // k1⊗k1 /64 * up^2
}

// =================================================================
// Kernel 2: 1x1 conv as WMMA GEMM (M=16 pixels/wave, N=16 (3 valid), K=512)
// + fused leaky_relu epilogue, adding the upsample already in d_out.
// 256 threads = 8 waves/block; wave w handles pixels [pix0, pix0+16).
// =================================================================
__global__ __launch_bounds__(256) void torgb_wmma_kernel(
    const float* __restrict__ x, const float* __restrict__ conv_w,
    const float* __restrict__ act_bias, const float* __restrict__ out_bias,
    float* __restrict__ out) {
  __shared__ float wlds[4 * LDSS];                 // rows 0..2 = scaled w, row 3 = zeros
  const float scale = 0.04419417382415922f;        // 1/sqrt(512)
  for (int i = threadIdx.x; i < 4 * LDSS; i += 256) {
    int row = i / LDSS;
    int k   = i - row * LDSS;
    wlds[i] = (row < 3 && k < CIN) ? conv_w[row * CIN + k] * scale : 0.f;
  }
  __syncthreads();

  int lane = threadIdx.x & 31;                     // wave32
  int wid  = blockIdx.x * 8 + (threadIdx.x >> 5);  // global wave id
  int b    = wid >> 12;                            // HW/16 = 4096 tiles per image
  int pix0 = (wid & 4095) << 4;                    // 16-pixel tile
  int hi   = lane >> 4;
  int m    = lane & 15;                            // pixel within tile (A rows)
  int n    = m;                                    // output channel (B cols)
  int row  = n < 3 ? n : 3;                        // n>=3 -> zero row (no divergence)
  const float* wrow = &wlds[row * LDSS];
  const float* xb   = x + (size_t)b * CIN * HW + pix0 + m;
  int koff = hi << 1;                              // K = k0 + 2*hi + vgpr

  v8f acc = {};
#pragma unroll 4
  for (int k0 = 0; k0 < CIN; k0 += 4) {
    int k = k0 + koff;
    v2f a, bf;
    a.x  = xb[(size_t)k * HW];                     // A: x[b, k,   pix0+m]
    a.y  = xb[(size_t)(k + 1) * HW];               // A: x[b, k+1, pix0+m]
    bf.x = wrow[k];                                // B: w_scaled[n, k]
    bf.y = wrow[k + 1];
    acc = __builtin_amdgcn_wmma_f32_16x16x4_f32(
        /*neg_a=*/false, a, /*neg_b=*/false, bf,
        /*c_mod=*/(short)0, acc, /*reuse_a=*/false, /*reuse_b=*/false);
  }

  if (n < 3) {
    float ab = act_bias[n];
    float ob = out_bias[n];
    // D layout: vgpr v, this lane -> pixel = pix0 + 8*hi + v  (8 contiguous)
    float* op = out + ((size_t)b * 3 + n) * HW + pix0 + (hi << 3);
    float4 u0 = *(const float4*)(op);
    float4 u1 = *(const float4*)(op + 4);
    float r[8];
#pragma unroll
    for (int v = 0; v < 8; ++v) {
      float av = acc[v] + ab;
      av = (av > 0.f ? av : av * 0.2f) * 1.4142135623730951f + ob;
      r[v] = av;
    }
    *(float4*)(op)     = make_float4(r[0] + u0.x, r[1] + u0.y, r[2] + u0.z, r[3] + u0.w);
    *(float4*)(op + 4) = make_float4(r[4] + u1.x, r[5] + u1.y, r[6] + u1.z, r[7] + u1.w);
  }
}

extern "C" void kernel_launch(void* const* d_in, const int* in_sizes, int n_in,
                              void* d_out, int out_size, void* d_ws, size_t ws_size,
                              hipStream_t stream) {
  const float* x        = (const float*)d_in[0];
  const float* skip     = (const float*)d_in[1];
  const float* conv_w   = (const float*)d_in[2];
  const float* act_bias = (const float*)d_in[3];
  const float* out_bias = (const float*)d_in[4];
  float* out = (float*)d_out;

  // 1) write upsampled skip into d_out
  int total_out = NB * 3 * HW;                     // 1,572,864
  upsample_blur_kernel<<<total_out / 256, 256, 0, stream>>>(skip, out);

  // 2) WMMA 1x1-conv + activation, accumulating onto d_out
  int nwaves = NB * (HW / 16);                     // 32,768 waves
  torgb_wmma_kernel<<<nwaves / 8, 256, 0, stream>>>(x, conv_w, act_bias, out_bias, out);
}